// Sparse2DLinear_36782099923515
// MI455X (gfx1250) — compile-verified
//
#include <hip/hip_runtime.h>
#include <hip/hip_bf16.h>

// Sparse 2D gather + dot:  out = sum_i coef[a_idx[i]*NUM_B + b_idx[i]] * values[i]
// Memory-latency bound: ~1M independent random f32 gathers from a 400MB table
// (does not fit 192MB L2).  Strategy:
//   * 4096 blocks x 256 threads (wave32) -> ~1M threads, one gather each, max MLP.
//   * non-temporal loads for the streaming idx/value arrays (preserve L2 for the table).
//   * wave reduction via V_WMMA_F32_16X16X4_F32 (B = ones -> row sums), then LDS
//     across the 8 waves, per-block partial to d_ws, second kernel reduces partials.
// Deterministic: fixed grid, no float atomics, fixed summation order.

#define NUM_B_DIM 10000

typedef __attribute__((ext_vector_type(2))) float v2f;
typedef __attribute__((ext_vector_type(8))) float v8f;

// Full-wave (32 lane) sum using the f32 WMMA unit.
// A (16x4) layout per ISA: lanes 0-15 VGPR0 = A[m,0], VGPR1 = A[m,1];
//                          lanes 16-31 VGPR0 = A[m,2], VGPR1 = A[m,3].
// Put each lane's accumulator in its VGPR0 slot, zero in VGPR1; B = all ones.
// Then D[m,n] = acc_m + acc_{m+16} (independent of n).  Lane L<16 holds
// D[0..7, L] in its 8 C VGPRs, lane L>=16 holds D[8..15, L-16]; summing the 8
// elements gives half-sums, xor-16 shuffle completes the wave sum in all lanes.
// REQUIRES: EXEC all ones (call from convergent code, full wave).
__device__ __forceinline__ float wave_sum_wmma(float acc) {
    v2f a;
    a.x = acc;
    a.y = 0.0f;
    v2f b;
    b.x = 1.0f;
    b.y = 1.0f;
    v8f c = {};
    c = __builtin_amdgcn_wmma_f32_16x16x4_f32(
        /*neg_a=*/false, a, /*neg_b=*/false, b,
        /*c_mod=*/(short)0, c, /*reuse_a=*/false, /*reuse_b=*/false);
    float half_sum = c[0] + c[1] + c[2] + c[3] + c[4] + c[5] + c[6] + c[7];
    // lanes 0-15 have sum(acc_0..acc_15-ish rows 0-7), lanes 16-31 rows 8-15.
    half_sum += __shfl_xor(half_sum, 16, 32);
    return half_sum;  // full wave sum, valid in every lane
}

__global__ void __launch_bounds__(256)
gather_dot_kernel(const int* __restrict__ a_idx,
                  const int* __restrict__ b_idx,
                  const float* __restrict__ values,
                  const float* __restrict__ coef,
                  float* __restrict__ partials,
                  int nnz) {
    const int tid    = blockIdx.x * blockDim.x + threadIdx.x;
    const int stride = gridDim.x * blockDim.x;

    float acc = 0.0f;
    // Grid-stride; with 1M threads each iteration count is ~1, all random
    // gathers are independent -> deep MLP across the whole device.
    for (int i = tid; i < nnz; i += stride) {
        // Streaming arrays: non-temporal so they don't evict coefficient
        // lines from the 192MB L2.
        const int   a = __builtin_nontemporal_load(a_idx + i);
        const int   bb = __builtin_nontemporal_load(b_idx + i);
        const float v = __builtin_nontemporal_load(values + i);
        // Random gather from the dense table (regular temporal: we WANT L2
        // to cache as much of the 400MB table as it can).
        const float g = coef[(size_t)a * NUM_B_DIM + (size_t)bb];
        acc = fmaf(g, v, acc);
    }

    // Wave sum on the matrix unit (free: kernel is memory bound).
    const float wsum = wave_sum_wmma(acc);

    __shared__ float smem[8];  // 256 threads / wave32 = 8 waves
    const int lane = threadIdx.x & 31;
    const int wave = threadIdx.x >> 5;
    if (lane == 0) smem[wave] = wsum;
    __syncthreads();

    if (threadIdx.x == 0) {
        float s = 0.0f;
        const int nwaves = blockDim.x >> 5;
        for (int w = 0; w < nwaves; ++w) s += smem[w];
        partials[blockIdx.x] = s;
    }
}

__global__ void __launch_bounds__(256)
reduce_partials_kernel(const float* __restrict__ partials,
                       int n,
                       float* __restrict__ out) {
    float acc = 0.0f;
    for (int i = threadIdx.x; i < n; i += blockDim.x) acc += partials[i];

    const float wsum = wave_sum_wmma(acc);

    __shared__ float smem[8];
    const int lane = threadIdx.x & 31;
    const int wave = threadIdx.x >> 5;
    if (lane == 0) smem[wave] = wsum;
    __syncthreads();

    if (threadIdx.x == 0) {
        float s = 0.0f;
        const int nwaves = blockDim.x >> 5;
        for (int w = 0; w < nwaves; ++w) s += smem[w];
        out[0] = s;
    }
}

extern "C" void kernel_launch(void* const* d_in, const int* in_sizes, int n_in,
                              void* d_out, int out_size, void* d_ws, size_t ws_size,
                              hipStream_t stream) {
    const int*   a_idx  = (const int*)d_in[0];
    const int*   b_idx  = (const int*)d_in[1];
    const float* values = (const float*)d_in[2];
    const float* coef   = (const float*)d_in[3];
    const int    nnz    = in_sizes[0];

    float* out      = (float*)d_out;
    float* partials = (float*)d_ws;

    // One block-partial per block in d_ws; cap by available scratch.
    int nblk = 4096;
    const int max_partials = (int)(ws_size / sizeof(float));
    if (nblk > max_partials) nblk = max_partials;
    if (nblk < 1) nblk = 1;

    gather_dot_kernel<<<nblk, 256, 0, stream>>>(a_idx, b_idx, values, coef,
                                                partials, nnz);
    reduce_partials_kernel<<<1, 256, 0, stream>>>(partials, nblk, out);
}